// NollaFraud_5239860101742
// MI455X (gfx1250) — compile-verified
//
#include <hip/hip_runtime.h>

#define N_NODES  50000
#define FEAT_DIM 32
#define EDIM     64
#define DEG      32
#define BATCH    512
#define N_REL    3

typedef __attribute__((ext_vector_type(2))) float v2f;
typedef __attribute__((ext_vector_type(8))) float v8f;

__device__ __forceinline__ int bcast_i(int x, int l) {
  return __builtin_amdgcn_readlane(x, l);
}
__device__ __forceinline__ float bcast_f(float x, int l) {
  return __int_as_float(__builtin_amdgcn_readlane(__float_as_int(x), l));
}

// ---------------- K0: row-softmax of alpha1 (128x3) and alpha2 (256x3) ---------
__global__ __launch_bounds__(384) void k_softmax(const float* __restrict__ a1,
                                                 const float* __restrict__ a2,
                                                 float* __restrict__ w1,
                                                 float* __restrict__ w2) {
  int t = threadIdx.x;
  const float* src; float* dst; int d;
  if (t < 2 * EDIM) { src = a1; dst = w1; d = t; }
  else              { src = a2; dst = w2; d = t - 2 * EDIM; }
  float x0 = src[d*3+0], x1 = src[d*3+1], x2 = src[d*3+2];
  float m  = fmaxf(x0, fmaxf(x1, x2));
  float e0 = expf(x0 - m), e1 = expf(x1 - m), e2 = expf(x2 - m);
  float inv = 1.0f / (e0 + e1 + e2);
  dst[d*3+0] = e0*inv; dst[d*3+1] = e1*inv; dst[d*3+2] = e2*inv;
}

// ---------------- K1: emb[50000,64] = feat[50000,32] @ w[32,64] + b (WMMA f32) --
// One wave per 16-row strip; 4 N-tiles of 16 cols; K=32 via 8x V_WMMA_F32_16X16X4_F32.
__global__ __launch_bounds__(256) void k_mlp(const float* __restrict__ feat,
                                             const float* __restrict__ w,
                                             const float* __restrict__ bias,
                                             float* __restrict__ emb) {
  const int lane = threadIdx.x & 31;
  const int wave = blockIdx.x * 8 + (threadIdx.x >> 5);
  const int m0 = wave * 16;
  if (m0 >= N_NODES) return;            // whole-wave exit; EXEC stays all-ones
  const int half = lane >> 4;           // K-half select (lanes 16-31 hold K+2,K+3)
  const int lrow = lane & 15;           // A row within tile / B,C column within tile
  v8f acc0 = {}, acc1 = {}, acc2 = {}, acc3 = {};
  const float* arow = feat + (size_t)(m0 + lrow) * FEAT_DIM;
  #pragma unroll
  for (int kb = 0; kb < FEAT_DIM; kb += 4) {
    const int ka = kb + 2 * half;
    v2f a; a.x = arow[ka]; a.y = arow[ka + 1];
    v2f b0, b1, b2, b3;
    b0.x = w[ka*EDIM +  0 + lrow]; b0.y = w[(ka+1)*EDIM +  0 + lrow];
    b1.x = w[ka*EDIM + 16 + lrow]; b1.y = w[(ka+1)*EDIM + 16 + lrow];
    b2.x = w[ka*EDIM + 32 + lrow]; b2.y = w[(ka+1)*EDIM + 32 + lrow];
    b3.x = w[ka*EDIM + 48 + lrow]; b3.y = w[(ka+1)*EDIM + 48 + lrow];
    acc0 = __builtin_amdgcn_wmma_f32_16x16x4_f32(false, a, false, b0, (short)0, acc0, false, false);
    acc1 = __builtin_amdgcn_wmma_f32_16x16x4_f32(false, a, false, b1, (short)0, acc1, false, false);
    acc2 = __builtin_amdgcn_wmma_f32_16x16x4_f32(false, a, false, b2, (short)0, acc2, false, false);
    acc3 = __builtin_amdgcn_wmma_f32_16x16x4_f32(false, a, false, b3, (short)0, acc3, false, false);
  }
  // C/D layout: VGPR v holds (row = m0 + v + 8*half, col = tile*16 + lrow)
  #pragma unroll
  for (int t = 0; t < 4; ++t) {
    const v8f acc = (t == 0) ? acc0 : (t == 1) ? acc1 : (t == 2) ? acc2 : acc3;
    const int col = t * 16 + lrow;
    const float bv = bias[col];
    #pragma unroll
    for (int v = 0; v < 8; ++v) {
      const int row = m0 + v + 8 * half;
      emb[(size_t)row * EDIM + col] = acc[v] + bv;
    }
  }
}

// ---------------- K2: agg1[50000,128] — layer-1 aggregate for every node -------
// One wave per node; lane owns dims {lane, lane+32} of 64-d vectors.
__global__ __launch_bounds__(256) void k_agg1(const int* __restrict__ adj,
                                              const float* __restrict__ emb,
                                              const float* __restrict__ wsm1,
                                              float* __restrict__ agg1) {
  const int lane = threadIdx.x & 31;
  const int node = blockIdx.x * 8 + (threadIdx.x >> 5);
  const float s0 = emb[(size_t)node * EDIM + lane];
  const float s1 = emb[(size_t)node * EDIM + lane + 32];
  float f1[N_REL][2];
  for (int r = 0; r < N_REL; ++r) {
    const int myid = adj[((size_t)r * N_NODES + node) * DEG + lane];
    int dup = 0;                               // duplicate of an earlier slot?
    for (int j = 0; j < DEG; ++j) {
      const int idj = bcast_i(myid, j);
      if (j < lane && idj == myid) dup = 1;
    }
    float a0 = 0.f, a1 = 0.f; int cnt = 0;
    for (int k = 0; k < DEG; ++k) {
      if (bcast_i(dup, k)) continue;           // scalar-uniform branch
      const int idk = bcast_i(myid, k);
      ++cnt;
      a0 += emb[(size_t)idk * EDIM + lane];
      a1 += emb[(size_t)idk * EDIM + lane + 32];
    }
    const float inv = 1.0f / (float)cnt;
    f1[r][0] = a0 * inv; f1[r][1] = a1 * inv;
  }
  #pragma unroll
  for (int q = 0; q < 4; ++q) {
    const int d = lane + 32 * q;
    float o = 0.f;
    #pragma unroll
    for (int r = 0; r < N_REL; ++r) {
      const float h = (q == 0) ? f1[r][0] : (q == 1) ? f1[r][1]
                    : (q == 2) ? (s0 - f1[r][0]) : (s1 - f1[r][1]);
      o += wsm1[d * 3 + r] * h;
    }
    agg1[(size_t)node * 2 * EDIM + d] = o;
  }
}

// ---------------- K3: batch pass — layer-2 aggregate + output head -------------
__global__ __launch_bounds__(256) void k_out(const int* __restrict__ nodes,
                                             const int* __restrict__ adj,
                                             const float* __restrict__ emb,
                                             const float* __restrict__ agg1,
                                             const float* __restrict__ wsm2,
                                             const float* __restrict__ w_out,
                                             const float* __restrict__ b_out,
                                             const float* __restrict__ prior,
                                             float* __restrict__ out) {
  const int lane = threadIdx.x & 31;
  const int b = blockIdx.x * 8 + (threadIdx.x >> 5);
  const int node = nodes[b];
  float s2[4];
  #pragma unroll
  for (int q = 0; q < 4; ++q) s2[q] = agg1[(size_t)node * 128 + lane + 32 * q];
  const float sf0 = emb[(size_t)node * EDIM + lane];
  const float sf1 = emb[(size_t)node * EDIM + lane + 32];
  float f1[N_REL][4];
  for (int r = 0; r < N_REL; ++r) {
    const int myid = adj[((size_t)r * N_NODES + node) * DEG + lane];
    int dup = 0;
    for (int j = 0; j < DEG; ++j) {
      const int idj = bcast_i(myid, j);
      if (j < lane && idj == myid) dup = 1;
    }
    float a[4] = {0.f, 0.f, 0.f, 0.f}; int cnt = 0;
    for (int k = 0; k < DEG; ++k) {
      if (bcast_i(dup, k)) continue;
      const int idk = bcast_i(myid, k);
      ++cnt;
      const float* p = agg1 + (size_t)idk * 128 + lane;
      a[0] += p[0]; a[1] += p[32]; a[2] += p[64]; a[3] += p[96];
    }
    const float inv = 1.0f / (float)cnt;
    #pragma unroll
    for (int q = 0; q < 4; ++q) f1[r][q] = a[q] * inv;
  }
  float p0 = 0.f, p1 = 0.f;
  // res[0:64] = self_f
  p0 += sf0 * w_out[lane*2+0] + sf1 * w_out[(lane+32)*2+0];
  p1 += sf0 * w_out[lane*2+1] + sf1 * w_out[(lane+32)*2+1];
  // res[64:192] = agg1(node)
  #pragma unroll
  for (int q = 0; q < 4; ++q) {
    const int d = 64 + lane + 32 * q;
    p0 += s2[q] * w_out[d*2+0];
    p1 += s2[q] * w_out[d*2+1];
  }
  // res[192:448] = agg2 = sum_r wsm2[:,r] * [f1_r, self2 - f1_r]
  #pragma unroll
  for (int q = 0; q < 8; ++q) {
    const int d = lane + 32 * q;
    float h = 0.f;
    #pragma unroll
    for (int r = 0; r < N_REL; ++r) {
      const float hr = (q < 4) ? f1[r][q] : (s2[q-4] - f1[r][q-4]);
      h += wsm2[d * 3 + r] * hr;
    }
    const int dd = 192 + d;
    p0 += h * w_out[dd*2+0];
    p1 += h * w_out[dd*2+1];
  }
  // wave32 reduction via readlane
  float t0 = 0.f, t1 = 0.f;
  for (int j = 0; j < 32; ++j) { t0 += bcast_f(p0, j); t1 += bcast_f(p1, j); }
  if (lane == 0) {
    out[b*2+0] = t0 + b_out[0] + logf(prior[0]);
    out[b*2+1] = t1 + b_out[1] + logf(prior[1]);
  }
}

extern "C" void kernel_launch(void* const* d_in, const int* in_sizes, int n_in,
                              void* d_out, int out_size, void* d_ws, size_t ws_size,
                              hipStream_t stream) {
  (void)in_sizes; (void)n_in; (void)out_size; (void)ws_size;
  const int*   nodes  = (const int*)  d_in[0];
  const float* feat   = (const float*)d_in[1];
  const int*   adj    = (const int*)  d_in[2];
  const float* w_mlp  = (const float*)d_in[3];
  const float* b_mlp  = (const float*)d_in[4];
  const float* alpha1 = (const float*)d_in[5];
  const float* alpha2 = (const float*)d_in[6];
  const float* w_out  = (const float*)d_in[7];
  const float* b_out  = (const float*)d_in[8];
  const float* prior  = (const float*)d_in[9];
  float* out = (float*)d_out;

  // workspace layout (floats): emb table, agg1 table, softmaxed alphas (~38.4 MB)
  float* emb  = (float*)d_ws;                          // 50000*64
  float* agg1 = emb  + (size_t)N_NODES * EDIM;         // 50000*128
  float* wsm1 = agg1 + (size_t)N_NODES * 2 * EDIM;     // 128*3
  float* wsm2 = wsm1 + 2 * EDIM * N_REL;               // 256*3

  k_softmax<<<1, 384, 0, stream>>>(alpha1, alpha2, wsm1, wsm2);
  k_mlp   <<<(N_NODES/16 + 7) / 8, 256, 0, stream>>>(feat, w_mlp, b_mlp, emb);
  k_agg1  <<<N_NODES / 8, 256, 0, stream>>>(adj, emb, wsm1, agg1);
  k_out   <<<BATCH / 8,   256, 0, stream>>>(nodes, adj, emb, agg1, wsm2, w_out, b_out, prior, out);
}